// GridEncoder_50646254354806
// MI455X (gfx1250) — compile-verified
//
#include <hip/hip_runtime.h>
#include <stdint.h>

#define NUM_LEVEL 16
#define LDS_L0_ENTRIES 4096u   // level-0 table entries (each 8 bytes) staged to LDS

// ---------------- compile-time level metadata (mirrors reference _grid_meta) ----------------
constexpr double kLevelScale = 1.38191288;

constexpr double cpow(double b, int e) { double r = 1.0; for (int i = 0; i < e; ++i) r *= b; return r; }
constexpr unsigned cceil_u(double x) {
  unsigned long long i = (unsigned long long)x;
  return (unsigned)(((double)i < x) ? (i + 1) : i);
}
constexpr unsigned chsize(int l) {
  unsigned r = cceil_u(16.0 * cpow(kLevelScale, l));
  unsigned long long c3 = (unsigned long long)r * r * r;
  unsigned long long t  = (c3 < (1ull << 19)) ? c3 : (1ull << 19);
  return (unsigned)((t + 7ull) & ~7ull);
}
constexpr unsigned coff(int l) { unsigned o = 0; for (int i = 0; i < l; ++i) o += chsize(i); return o; }
constexpr float    cscale(int l) { return (float)(16.0 * cpow(kLevelScale, l) - 1.0); }
constexpr unsigned cres(int l) { return cceil_u(16.0 * cpow(kLevelScale, l) - 1.0) + 1u; }
constexpr unsigned chashed(int l) {
  unsigned r = cres(l);
  return ((unsigned long long)r * r * r > (unsigned long long)chsize(l)) ? 1u : 0u;
}

#define TAB(F) { F(0),F(1),F(2),F(3),F(4),F(5),F(6),F(7),F(8),F(9),F(10),F(11),F(12),F(13),F(14),F(15) }
__constant__ unsigned c_off[NUM_LEVEL]    = TAB(coff);
__constant__ unsigned c_res[NUM_LEVEL]    = TAB(cres);
__constant__ unsigned c_hashed[NUM_LEVEL] = TAB(chashed);
__constant__ float    c_scale[NUM_LEVEL]  = TAB(cscale);
#undef TAB

typedef unsigned int u32x4 __attribute__((ext_vector_type(4)));
typedef int          i32x4 __attribute__((ext_vector_type(4)));
typedef int          i32x8 __attribute__((ext_vector_type(8)));

// One thread per (point, level). lane&15 == level so 16 consecutive lanes
// write one contiguous 128B output row chunk (coalesced b64 stores, wave32).
__global__ __launch_bounds__(256) void grid_encode_kernel(
    const float* __restrict__ means,   // (N,3)
    const float* __restrict__ emb,     // (offs[-1], 2)
    float* __restrict__ out,           // (N, 32)
    int npts)
{
  extern __shared__ float2 s_l0[];     // 4096 * 8B = 32 KB: level-0 table

  // ---- Stage level-0 embedding table into LDS ----
#if __has_builtin(__builtin_amdgcn_tensor_load_to_lds)
  if (threadIdx.x < 32u) {             // wave 0 issues the TDM DMA once
    unsigned long long ga = (unsigned long long)(uintptr_t)emb;     // tile start = level 0 start
    unsigned lds_base = (unsigned)(uintptr_t)(&s_l0[0]);            // LDS byte offset
    // D# group 0: count=1 | lds_addr | global_addr[56:0] | type=2
    u32x4 g0 = { 1u,
                 lds_base,
                 (unsigned)(ga & 0xffffffffu),
                 (unsigned)((ga >> 32) & 0x01ffffffu) | (2u << 30) };
    // D# group 1: data_size=3 (8B elems); tensor_dim0=4096; tensor_dim1=1;
    //             tile_dim0=4096; tile_dim1=1; tensor_dim0_stride=4096
    i32x8 g1 = { (int)(3u << 16),
                 (int)((LDS_L0_ENTRIES & 0xffffu) << 16),   // tensor_dim0[15:0] @ bits63:48
                 (int)(1u << 16),                           // tensor_dim0 hi=0 | tensor_dim1=1
                 (int)((LDS_L0_ENTRIES & 0xffffu) << 16),   // tensor_dim1 hi=0 | tile_dim0=4096
                 1,                                         // tile_dim1=1, tile_dim2=0
                 (int)LDS_L0_ENTRIES,                       // tensor_dim0_stride[31:0]
                 0, 0 };
    i32x4 gz = { 0, 0, 0, 0 };
#if defined(__clang_major__) && (__clang_major__ >= 23)
    i32x8 gz8 = { 0, 0, 0, 0, 0, 0, 0, 0 };
    __builtin_amdgcn_tensor_load_to_lds(g0, g1, gz, gz, gz8, 0);
#else
    __builtin_amdgcn_tensor_load_to_lds(g0, g1, gz, gz, 0);
#endif
#if __has_builtin(__builtin_amdgcn_s_wait_tensorcnt)
    __builtin_amdgcn_s_wait_tensorcnt(0);
#else
    asm volatile("s_wait_tensorcnt 0x0" ::: "memory");
#endif
  }
#else
  { // fallback: cooperative copy (host pass / toolchains without the TDM builtin)
    const float2* ef2 = (const float2*)emb;
    for (unsigned i = threadIdx.x; i < LDS_L0_ENTRIES; i += blockDim.x) s_l0[i] = ef2[i];
  }
#endif
  __syncthreads();

  unsigned t = blockIdx.x * 256u + threadIdx.x;
  unsigned n = t >> 4;
  unsigned l = t & 15u;
  if (n >= (unsigned)npts) return;

  float mx = means[3u * n + 0u];
  float my = means[3u * n + 1u];
  float mz = means[3u * n + 2u];

  float sc = c_scale[l];
  float x0 = (mx + 1.0f) * 0.5f;
  float y0 = (my + 1.0f) * 0.5f;
  float z0 = (mz + 1.0f) * 0.5f;
  float px = x0 * sc, py = y0 * sc, pz = z0 * sc;
  float gx = floorf(px), gy = floorf(py), gz = floorf(pz);
  float fx = px - gx, fy = py - gy, fz = pz - gz;
  unsigned ix = (unsigned)gx, iy = (unsigned)gy, iz = (unsigned)gz;

  unsigned off = c_off[l];
  unsigned res = c_res[l];
  bool hashed  = (c_hashed[l] != 0u);

  // All 8 corner indices first (hash levels: hsize == 2^19 -> mask, dense: provably in range)
  unsigned id[8];
#pragma unroll
  for (int c = 0; c < 8; ++c) {
    unsigned cx = ix + (unsigned)(c & 1);
    unsigned cy = iy + (unsigned)((c >> 1) & 1);
    unsigned cz = iz + (unsigned)((c >> 2) & 1);
    unsigned h  = cx ^ (cy * 2654435761u) ^ (cz * 805459861u);
    unsigned d  = cx + cy * res + cz * res * res;
    id[c] = off + (hashed ? (h & 524287u) : d);
  }

  // Issue all 8 gathers back-to-back so they clause & overlap.
  float2 f[8];
  if (l == 0u) {
#pragma unroll
    for (int c = 0; c < 8; ++c) f[c] = s_l0[id[c]];        // LDS-resident level 0
  } else {
    const float2* ef2 = (const float2*)emb;
#pragma unroll
    for (int c = 0; c < 8; ++c) f[c] = ef2[id[c]];         // L2-resident gathers
  }

  float wx[2] = { 1.0f - fx, fx };
  float wy[2] = { 1.0f - fy, fy };
  float wz[2] = { 1.0f - fz, fz };
  float ax = 0.0f, ay = 0.0f;
#pragma unroll
  for (int c = 0; c < 8; ++c) {
    float w = wx[c & 1] * wy[(c >> 1) & 1] * wz[(c >> 2) & 1];
    ax = fmaf(w, f[c].x, ax);
    ay = fmaf(w, f[c].y, ay);
  }

  float2 r; r.x = ax; r.y = ay;
  ((float2*)out)[(size_t)n * 16u + l] = r;   // out[n, 2l:2l+2], coalesced across lanes
}

extern "C" void kernel_launch(void* const* d_in, const int* in_sizes, int n_in,
                              void* d_out, int out_size, void* d_ws, size_t ws_size,
                              hipStream_t stream) {
  (void)n_in; (void)out_size; (void)d_ws; (void)ws_size;
  const float* means = (const float*)d_in[0];
  const float* emb   = (const float*)d_in[1];
  float* out         = (float*)d_out;

  int npts   = in_sizes[0] / 3;
  int total  = npts * NUM_LEVEL;
  int threads = 256;
  int blocks  = (total + threads - 1) / threads;
  size_t shmem = (size_t)LDS_L0_ENTRIES * sizeof(float2);  // 32 KB dynamic LDS

  grid_encode_kernel<<<blocks, threads, shmem, stream>>>(means, emb, out, npts);
}